// spatialBlock_4054449127575
// MI455X (gfx1250) — compile-verified
//
#include <hip/hip_runtime.h>
#include <math.h>

// Problem constants (match the reference)
#define GW   4        // WINDOW
#define GN   50000    // nodes
#define GE   400000   // edges
#define GD   128      // feature dim

#define SCAN_T   256
#define NBLK     ((GN + SCAN_T - 1) / SCAN_T)      // 196 scan blocks

typedef float v2f __attribute__((ext_vector_type(2)));
typedef float v8f __attribute__((ext_vector_type(8)));

// ---------------------------------------------------------------------------
// GEMM: C[N x 128] = A[N x 128] @ W[128 x 128] via V_WMMA_F32_16X16X4_F32.
// Block = 256 threads = 8 waves. W staged once into LDS per block.
// Each wave computes a 16x128 strip (8 accumulators); each A float2 is
// reused across 8 WMMAs. B operands pre-loaded into 8 registers per k-step
// so LDS waits can overlap the WMMA chain.
//
// LDS layout: K-pair interleaved, pitch 288 dwords -> lanes 0-15 (pair p)
// and 16-31 (pair p+1) hit disjoint 32-bank halves: conflict-free b64.
// ---------------------------------------------------------------------------
#define LDS_PITCH 288                       // dwords per K-pair row
#define LDS_DW    (64 * LDS_PITCH)          // 18432 dwords = 73728 B

__global__ __launch_bounds__(256) void gemm_f32_wmma(
    const float* __restrict__ A,      // [GN x GD]
    const float* __restrict__ Wm,     // [GD x GD]
    float* __restrict__ C)            // [GN x GD]
{
  extern __shared__ float lds[];      // LDS_DW floats (dynamic: 73728 bytes)

  const int tid = threadIdx.x;
  for (int i = tid; i < GD * GD; i += 256) {
    const int k = i >> 7;
    const int n = i & 127;
    lds[(k >> 1) * LDS_PITCH + n * 2 + (k & 1)] = Wm[i];
  }
  __syncthreads();

  const int lane   = tid & 31;
  const int wv     = tid >> 5;
  const int stripe = blockIdx.x * 8 + wv;     // 16-row stripe id
  const int m0     = stripe << 4;
  const int ml     = lane & 15;
  const int hi     = lane >> 4;               // 0 or 1
  const bool active = (m0 < GN);

  const float* __restrict__ arow =
      A + (size_t)((active ? m0 : 0) + ml) * GD;

  v8f acc[8];
#pragma unroll
  for (int t = 0; t < 8; ++t) acc[t] = (v8f)0.f;

#pragma unroll 2
  for (int k0 = 0; k0 < GD; k0 += 4) {
    const int ka = k0 + 2 * hi;
    const v2f a = *(const v2f*)(arow + ka);             // contiguous, 8B
    const float* bbase = lds + (ka >> 1) * LDS_PITCH;
    v2f b[8];
#pragma unroll
    for (int t = 0; t < 8; ++t)
      b[t] = *(const v2f*)(bbase + (t * 16 + ml) * 2);  // ds_load_b64 x8
#pragma unroll
    for (int t = 0; t < 8; ++t)
      acc[t] = __builtin_amdgcn_wmma_f32_16x16x4_f32(
          false, a, false, b[t], (short)0, acc[t], false, false);
  }

  if (active) {
#pragma unroll
    for (int t = 0; t < 8; ++t) {
#pragma unroll
      for (int r = 0; r < 8; ++r) {
        C[(size_t)(m0 + r + 8 * hi) * GD + t * 16 + ml] = acc[t][r];
      }
    }
  }
}

// ---------------------------------------------------------------------------
// Batched CSR build (all 4 windows at once, blockIdx.y = window).
// Counting sort of edges by destination; emits a dst-sorted packed
// {src, weight} int2 stream so the aggregate loop is a single 8B load
// per edge instead of a 3-deep pointer chase.
// ---------------------------------------------------------------------------
__global__ __launch_bounds__(256) void k_zero(int* __restrict__ deg4) {
  const int i = blockIdx.x * 256 + threadIdx.x;
  if (i < GN) deg4[(size_t)blockIdx.y * GN + i] = 0;
}

__global__ __launch_bounds__(256) void k_hist(const int* __restrict__ ei,
                                              int* __restrict__ deg4) {
  const int e = blockIdx.x * 256 + threadIdx.x;
  const int w = blockIdx.y;
  if (e < GE) {
    const int d = ei[(size_t)w * 2 * GE + GE + e];
    atomicAdd(&deg4[(size_t)w * GN + d], 1);
  }
}

__global__ __launch_bounds__(256) void k_scan1(const int* __restrict__ deg4,
                                               int* __restrict__ offs4,
                                               int* __restrict__ blk4) {
  __shared__ int s[SCAN_T];
  const int t = threadIdx.x;
  const int i = blockIdx.x * SCAN_T + t;
  const int w = blockIdx.y;
  const int v = (i < GN) ? deg4[(size_t)w * GN + i] : 0;
  s[t] = v;
  __syncthreads();
  for (int off = 1; off < SCAN_T; off <<= 1) {
    const int x = (t >= off) ? s[t - off] : 0;
    __syncthreads();
    s[t] += x;
    __syncthreads();
  }
  if (i < GN) offs4[(size_t)w * (GN + 1) + i] = s[t] - v;   // excl in block
  if (t == SCAN_T - 1) blk4[w * 256 + blockIdx.x] = s[t];
}

__global__ __launch_bounds__(256) void k_scan2(int* __restrict__ blk4) {
  __shared__ int s[SCAN_T];
  const int t = threadIdx.x;
  const int w = blockIdx.y;
  const int v = (t < NBLK) ? blk4[w * 256 + t] : 0;
  s[t] = v;
  __syncthreads();
  for (int off = 1; off < SCAN_T; off <<= 1) {
    const int x = (t >= off) ? s[t - off] : 0;
    __syncthreads();
    s[t] += x;
    __syncthreads();
  }
  if (t < NBLK) blk4[w * 256 + t] = s[t] - v;               // excl block offs
}

__global__ __launch_bounds__(256) void k_scan3(int* __restrict__ offs4,
                                               const int* __restrict__ blk4,
                                               int* __restrict__ cursor4) {
  const int t = threadIdx.x;
  const int i = blockIdx.x * SCAN_T + t;
  const int w = blockIdx.y;
  if (i < GN) {
    const int o = offs4[(size_t)w * (GN + 1) + i] + blk4[w * 256 + blockIdx.x];
    offs4[(size_t)w * (GN + 1) + i] = o;
    cursor4[(size_t)w * GN + i] = o;
  }
  if (i == 0) offs4[(size_t)w * (GN + 1) + GN] = GE;
}

__global__ __launch_bounds__(256) void k_place(const int* __restrict__ ei,
                                               const float* __restrict__ ew,
                                               int* __restrict__ cursor4,
                                               int2* __restrict__ pedge4) {
  const int e = blockIdx.x * 256 + threadIdx.x;
  const int w = blockIdx.y;
  if (e < GE) {
    const int src = ei[(size_t)w * 2 * GE + e];
    const int dst = ei[(size_t)w * 2 * GE + GE + e];
    const float wgt = ew[(size_t)w * GE + e];
    const int p = atomicAdd(&cursor4[(size_t)w * GN + dst], 1);
    pedge4[(size_t)w * GE + p] = make_int2(src, __float_as_int(wgt));
  }
}

// ---------------------------------------------------------------------------
// Aggregation with fused bias + ELU: one wave per node, full 128-float row
// in registers (float4/lane). Inner loop: one 8B packed-metadata load ->
// one 512B coalesced row gather; unrolled x2 to keep two gathers in flight.
// out[n] = elu( bias + sum_{e in CSR[n]} ew[e] * h[src[e]] )
// ---------------------------------------------------------------------------
__global__ __launch_bounds__(256) void k_aggregate(
    const float* __restrict__ h,       // [GN x GD]
    const int2*  __restrict__ pedge,   // [GE] dst-sorted {src, wgt}
    const int*   __restrict__ offs,    // [GN+1]
    const float* __restrict__ bias,    // [GD]
    float*       __restrict__ out)     // [GN x GD]
{
  const int n = blockIdx.x * 8 + (threadIdx.x >> 5);
  if (n >= GN) return;
  const int lane = threadIdx.x & 31;

  float4 acc = ((const float4*)bias)[lane];
  const int beg = offs[n];
  const int end = offs[n + 1];

  int j = beg;
  for (; j + 2 <= end; j += 2) {
    const int2 p0 = pedge[j];
    const int2 p1 = pedge[j + 1];
    const float w0 = __int_as_float(p0.y);
    const float w1 = __int_as_float(p1.y);
    const float4 v0 = *(const float4*)(h + (size_t)p0.x * GD + (lane << 2));
    const float4 v1 = *(const float4*)(h + (size_t)p1.x * GD + (lane << 2));
    acc.x = fmaf(w0, v0.x, acc.x); acc.y = fmaf(w0, v0.y, acc.y);
    acc.z = fmaf(w0, v0.z, acc.z); acc.w = fmaf(w0, v0.w, acc.w);
    acc.x = fmaf(w1, v1.x, acc.x); acc.y = fmaf(w1, v1.y, acc.y);
    acc.z = fmaf(w1, v1.z, acc.z); acc.w = fmaf(w1, v1.w, acc.w);
  }
  if (j < end) {
    const int2 p0 = pedge[j];
    const float w0 = __int_as_float(p0.y);
    const float4 v0 = *(const float4*)(h + (size_t)p0.x * GD + (lane << 2));
    acc.x = fmaf(w0, v0.x, acc.x); acc.y = fmaf(w0, v0.y, acc.y);
    acc.z = fmaf(w0, v0.z, acc.z); acc.w = fmaf(w0, v0.w, acc.w);
  }

  acc.x = acc.x > 0.f ? acc.x : expm1f(acc.x);
  acc.y = acc.y > 0.f ? acc.y : expm1f(acc.y);
  acc.z = acc.z > 0.f ? acc.z : expm1f(acc.z);
  acc.w = acc.w > 0.f ? acc.w : expm1f(acc.w);
  ((float4*)(out + (size_t)n * GD))[lane] = acc;
}

// ---------------------------------------------------------------------------
extern "C" void kernel_launch(void* const* d_in, const int* in_sizes, int n_in,
                              void* d_out, int out_size, void* d_ws, size_t ws_size,
                              hipStream_t stream) {
  const float* x  = (const float*)d_in[0];   // [W,N,D]
  const int*   ei = (const int*)  d_in[1];   // [W,2,E]
  const float* ew = (const float*)d_in[2];   // [W,E]
  const float* W0 = (const float*)d_in[3];   // [W,D,D]
  const float* b0 = (const float*)d_in[4];   // [W,D]
  const float* W1 = (const float*)d_in[5];   // [W,D,D]
  const float* b1 = (const float*)d_in[6];   // [W,D]
  float* out = (float*)d_out;                // [W,N,D]

  // Workspace layout
  float* bufA   = (float*)d_ws;                       // [N,D]
  float* bufB   = bufA + (size_t)GN * GD;             // [N,D]
  int*   deg4    = (int*)(bufB + (size_t)GN * GD);    // [4][N]
  int*   offs4   = deg4 + (size_t)4 * GN;             // [4][N+1]
  int*   cursor4 = offs4 + (size_t)4 * (GN + 1);      // [4][N]
  int*   blk4    = cursor4 + (size_t)4 * GN;          // [4][256]
  int2*  pedge4  = (int2*)(blk4 + 4 * 256);           // [4][E]

  const dim3 blk(256);
  const dim3 gemm_grid((GN / 16 + 7) / 8);            // 391 (8 stripes/block)
  const dim3 node_grid(NBLK, GW);                     // 196 x 4
  const dim3 edge_grid((GE + 255) / 256, GW);         // 1563 x 4
  const dim3 one_grid(1, GW);
  const dim3 agg_grid((GN + 7) / 8);                  // 6250
  const size_t gemm_lds = LDS_DW * sizeof(float);     // 73728 B

  // ---- CSR build for all 4 windows (batched) ----
  k_zero <<<node_grid, blk, 0, stream>>>(deg4);
  k_hist <<<edge_grid, blk, 0, stream>>>(ei, deg4);
  k_scan1<<<node_grid, blk, 0, stream>>>(deg4, offs4, blk4);
  k_scan2<<<one_grid,  blk, 0, stream>>>(blk4);
  k_scan3<<<node_grid, blk, 0, stream>>>(offs4, blk4, cursor4);
  k_place<<<edge_grid, blk, 0, stream>>>(ei, ew, cursor4, pedge4);

  for (int w = 0; w < GW; ++w) {
    const float* xw  = x  + (size_t)w * GN * GD;
    const float* W0w = W0 + (size_t)w * GD * GD;
    const float* W1w = W1 + (size_t)w * GD * GD;
    const int*   offs  = offs4  + (size_t)w * (GN + 1);
    const int2*  pedge = pedge4 + (size_t)w * GE;
    float* outw = out + (size_t)w * GN * GD;

    // ---- conv0: bufB = elu( agg(x@W0) + b0 ) ----
    gemm_f32_wmma<<<gemm_grid, blk, gemm_lds, stream>>>(xw, W0w, bufA);
    k_aggregate  <<<agg_grid,  blk, 0,        stream>>>(bufA, pedge, offs,
                                                        b0 + (size_t)w * GD, bufB);

    // ---- conv1: out = elu( agg(bufB@W1) + b1 ) ----
    gemm_f32_wmma<<<gemm_grid, blk, gemm_lds, stream>>>(bufB, W1w, bufA);
    k_aggregate  <<<agg_grid,  blk, 0,        stream>>>(bufA, pedge, offs,
                                                        b1 + (size_t)w * GD, outw);
  }
}